// DoseOptimizationLoss_5437428597203
// MI455X (gfx1250) — compile-verified
//
#include <hip/hip_runtime.h>
#include <math.h>

typedef __attribute__((ext_vector_type(2))) float v2f;
typedef __attribute__((ext_vector_type(8))) float v8f;

#define VOLX 160
#define NVOX (VOLX * VOLX * VOLX)
#define NSEEDS 32
#define NTILES (NVOX / 16)
#define TPR 10   /* tiles per k-row: 160/16 */
#define NEG_LOG2E (-1.44269504088896340736f)

// ---------------- init: zero the 5 accumulators -------------------------
__global__ void dose_init(float* __restrict__ acc) {
    if (threadIdx.x < 8) acc[threadIdx.x] = 0.0f;
}

// ---------------- seed prep: 10 poly coeffs per seed --------------------
// Bmat layout: row-major [K=12][S=32], coords centered at 80.
// All coefficients are pre-scaled by -log2(e), so the WMMA output is
// -log2(e)*quad and feeds v_exp_f32 (native exp2) with no extra VALU.
__global__ void dose_seed_prep(const float* __restrict__ x, float* __restrict__ Bmat) {
    int s = threadIdx.x;
    if (s >= NSEEDS) return;

    float cx = x[6 * s + 0] * 160.0f - 80.0f;
    float cy = x[6 * s + 1] * 160.0f - 80.0f;
    float cz = x[6 * s + 2] * 160.0f - 80.0f;
    float dx = x[6 * s + 3];
    float dy = x[6 * s + 4];
    float dz = x[6 * s + 5];

    // Rodrigues: rotate e_x onto d. angle = acos(clip(dx)), axis = norm(0,-dz,dy)
    float dc = fminf(0.999999f, fmaxf(-0.999999f, dx));
    float ct = dc;
    float st = sqrtf(fmaxf(0.0f, 1.0f - dc * dc));
    float ay = -dz, az = dy;
    float nrm = sqrtf(ay * ay + az * az);
    float den = (nrm > 1e-8f) ? nrm : 1.0f;
    float a0 = 0.0f, a1 = ay / den, a2 = az / den;

    float a[3] = {a0, a1, a2};
    float Km[3][3] = {{0.0f, -a2, a1}, {a2, 0.0f, -a0}, {-a1, a0, 0.0f}};
    float R[3][3];
    bool use_id = fabsf(dx) >= 0.99f;
#pragma unroll
    for (int i = 0; i < 3; ++i)
#pragma unroll
        for (int j = 0; j < 3; ++j) {
            float eye = (i == j) ? 1.0f : 0.0f;
            float r = ct * eye + (1.0f - ct) * a[i] * a[j] + st * Km[i][j];
            R[i][j] = use_id ? eye : r;
        }

    // A = R^T D R, D = diag(1/(2*sigma^2)), sigma = (8,4,4)
    const float dv[3] = {1.0f / 128.0f, 1.0f / 32.0f, 1.0f / 32.0f};
    float A[3][3];
#pragma unroll
    for (int i = 0; i < 3; ++i)
#pragma unroll
        for (int j = 0; j < 3; ++j) {
            float acc = 0.0f;
#pragma unroll
            for (int k = 0; k < 3; ++k) acc += dv[k] * R[k][i] * R[k][j];
            A[i][j] = acc;
        }

    float c[3] = {cx, cy, cz};
    float Ac[3];
#pragma unroll
    for (int i = 0; i < 3; ++i)
        Ac[i] = A[i][0] * c[0] + A[i][1] * c[1] + A[i][2] * c[2];
    float cAc = c[0] * Ac[0] + c[1] * Ac[1] + c[2] * Ac[2];

    // quad(v') = cAc - 2 Ac.v' + v'^T A v'  over features [1,x,y,z,x2,y2,z2,xy,xz,yz,0,0]
    const float g = NEG_LOG2E;     // fold -log2(e) into every coefficient
    Bmat[0 * 32 + s]  = g * cAc;
    Bmat[1 * 32 + s]  = g * -2.0f * Ac[0];
    Bmat[2 * 32 + s]  = g * -2.0f * Ac[1];
    Bmat[3 * 32 + s]  = g * -2.0f * Ac[2];
    Bmat[4 * 32 + s]  = g * A[0][0];
    Bmat[5 * 32 + s]  = g * A[1][1];
    Bmat[6 * 32 + s]  = g * A[2][2];
    Bmat[7 * 32 + s]  = g * 2.0f * A[0][1];
    Bmat[8 * 32 + s]  = g * 2.0f * A[0][2];
    Bmat[9 * 32 + s]  = g * 2.0f * A[1][2];
    Bmat[10 * 32 + s] = 0.0f;
    Bmat[11 * 32 + s] = 0.0f;
}

// Fold-reduce: merge two partial-sum arrays per butterfly stage.
// Lanes with (lane & m)==0 end up carrying array `a` sums, others array `b`.
__device__ __forceinline__ float fold(float a, float b, int m, int lane) {
    float sel = (lane & m) ? a : b;      // value sent to partner
    float t = __shfl_xor(sel, m);
    return ((lane & m) ? b : a) + t;
}

// ---------------- main kernel: WMMA quad forms + exp2 + reductions ------
__global__ void __launch_bounds__(256) dose_main(const float* __restrict__ rv,
                                                 const float* __restrict__ om,
                                                 const float* __restrict__ Bmat,
                                                 float* __restrict__ acc) {
    const int lane = threadIdx.x & 31;
    const int hi = lane >> 4;       // 0: K even-pair lanes, 1: K odd-pair lanes
    const int l15 = lane & 15;
    const int waveId = (blockIdx.x * blockDim.x + threadIdx.x) >> 5;
    const int nWaves = (gridDim.x * blockDim.x) >> 5;

    // B fragments (constant across tiles). 4x16 f32 B layout:
    // VGPR0: lanes0-15 K=0, lanes16-31 K=2 ; VGPR1: K=1 / K=3.
    const int r0 = 2 * hi, r1 = 2 * hi + 1;
    const int n0 = l15, n1 = 16 + l15;
    v2f b0t0, b1t0, b2t0, b0t1, b1t1, b2t1;
    b0t0.x = Bmat[(0 + r0) * 32 + n0]; b0t0.y = Bmat[(0 + r1) * 32 + n0];
    b1t0.x = Bmat[(4 + r0) * 32 + n0]; b1t0.y = Bmat[(4 + r1) * 32 + n0];
    b2t0.x = Bmat[(8 + r0) * 32 + n0]; b2t0.y = Bmat[(8 + r1) * 32 + n0];
    b0t1.x = Bmat[(0 + r0) * 32 + n1]; b0t1.y = Bmat[(0 + r1) * 32 + n1];
    b1t1.x = Bmat[(4 + r0) * 32 + n1]; b1t1.y = Bmat[(4 + r1) * 32 + n1];
    b2t1.x = Bmat[(8 + r0) * 32 + n1]; b2t1.y = Bmat[(8 + r1) * 32 + n1];

    float acc_rv = 0.0f, acc_se = 0.0f, acc_so = 0.0f, acc_eff = 0.0f, acc_rad = 0.0f;

    for (int tile = waveId; tile < NTILES; tile += nWaves) {
        // Tile decomposition is wave-uniform: all 16 voxels share i and j
        // (160 % 16 == 0); only k varies by lane. Force it onto the SALU.
        int t = __builtin_amdgcn_readfirstlane(tile);
        int iq = t / (TPR * VOLX);
        int rem = t - iq * (TPR * VOLX);
        int jq = rem / TPR;
        int kt = rem - jq * TPR;                 // tile index within the k-row

        float xf = (float)iq - 80.0f;            // wave-uniform
        float yf = (float)jq - 80.0f;            // wave-uniform
        float zf = (float)(kt * 16 + l15) - 80.0f;

        // A fragments (16x4 f32 layout: lanes0-15 hold K=0,1; lanes16-31 K=2,3)
        // features: [1, x, y, z, x2, y2, z2, xy, xz, yz, 0, 0]
        // Branchless per-component selects keep EXEC untouched around WMMA.
        v2f a0, a1, a2;
        a0.x = hi ? yf : 1.0f;
        a0.y = hi ? zf : xf;
        a1.x = hi ? zf * zf : xf * xf;
        a1.y = hi ? xf * yf : yf * yf;
        a2.x = hi ? 0.0f : xf * zf;
        a2.y = hi ? 0.0f : yf * zf;

        v8f c0 = {};
        v8f c1 = {};
        c0 = __builtin_amdgcn_wmma_f32_16x16x4_f32(false, a0, false, b0t0, (short)0, c0, false, false);
        c0 = __builtin_amdgcn_wmma_f32_16x16x4_f32(false, a1, false, b1t0, (short)0, c0, false, false);
        c0 = __builtin_amdgcn_wmma_f32_16x16x4_f32(false, a2, false, b2t0, (short)0, c0, false, false);
        c1 = __builtin_amdgcn_wmma_f32_16x16x4_f32(false, a0, false, b0t1, (short)0, c1, false, false);
        c1 = __builtin_amdgcn_wmma_f32_16x16x4_f32(false, a1, false, b1t1, (short)0, c1, false, false);
        c1 = __builtin_amdgcn_wmma_f32_16x16x4_f32(false, a2, false, b2t1, (short)0, c1, false, false);

        // WMMA output is already -log2(e)*quad: feed native exp2 directly.
        float s0 = __builtin_amdgcn_exp2f(c0[0]) + __builtin_amdgcn_exp2f(c1[0]);
        float s1 = __builtin_amdgcn_exp2f(c0[1]) + __builtin_amdgcn_exp2f(c1[1]);
        float s2 = __builtin_amdgcn_exp2f(c0[2]) + __builtin_amdgcn_exp2f(c1[2]);
        float s3 = __builtin_amdgcn_exp2f(c0[3]) + __builtin_amdgcn_exp2f(c1[3]);
        float s4 = __builtin_amdgcn_exp2f(c0[4]) + __builtin_amdgcn_exp2f(c1[4]);
        float s5 = __builtin_amdgcn_exp2f(c0[5]) + __builtin_amdgcn_exp2f(c1[5]);
        float s6 = __builtin_amdgcn_exp2f(c0[6]) + __builtin_amdgcn_exp2f(c1[6]);
        float s7 = __builtin_amdgcn_exp2f(c0[7]) + __builtin_amdgcn_exp2f(c1[7]);

        // Fold-reduction over the 16-lane seed dimension: 8 shuffles total.
        // Result: lane l (mod 16) carries radiation of row ((l&7) + 8*hi).
        float u0 = fold(s0, s1, 1, lane);
        float u1 = fold(s2, s3, 1, lane);
        float u2 = fold(s4, s5, 1, lane);
        float u3 = fold(s6, s7, 1, lane);
        float w0 = fold(u0, u1, 2, lane);
        float w1 = fold(u2, u3, 2, lane);
        float z  = fold(w0, w1, 4, lane);
        z += __shfl_xor(z, 8);

        if (l15 < 8) {
            float r = z;                              // radiation of this voxel
            int vm = t * 16 + hi * 8 + (lane & 7);
            float fv = rv[vm];
            float fo = om[vm];
            float eff = r * fv;
            float outd = r * fo;
            acc_rv  += fv;
            acc_se  += 1.0f / (1.0f + __expf(1.0f - eff));            // sigmoid(eff - 1)
            acc_so  += 1.0f / (1.0f + __expf(100.0f * (0.5f - outd))); // sigmoid(100*(out-0.5))
            acc_eff += eff;
            acc_rad += r;
        }
    }

    // full-wave reduction of the 5 accumulators, then one atomic per wave
#pragma unroll
    for (int d = 1; d < 32; d <<= 1) {
        acc_rv  += __shfl_xor(acc_rv, d);
        acc_se  += __shfl_xor(acc_se, d);
        acc_so  += __shfl_xor(acc_so, d);
        acc_eff += __shfl_xor(acc_eff, d);
        acc_rad += __shfl_xor(acc_rad, d);
    }
    if (lane == 0) {
        atomicAdd(&acc[0], acc_rv);
        atomicAdd(&acc[1], acc_se);
        atomicAdd(&acc[2], acc_so);
        atomicAdd(&acc[3], acc_eff);
        atomicAdd(&acc[4], acc_rad);
    }
}

// ---------------- finalize: combine sums into the scalar loss -----------
__global__ void dose_finalize(const float* __restrict__ acc, float* __restrict__ out) {
    float num_target = acc[0];
    float eff_units  = acc[1];
    float sig_out    = acc[2];
    float sum_eff    = acc[3];
    float sum_rad    = acc[4];
    float cur_dvh = eff_units / num_target;
    float margin_penalty = 0.9f - cur_dvh;
    float unsafe_rate = sig_out / num_target;
    float overdose_penalty = 1.0f - sum_eff / sum_rad;
    out[0] = margin_penalty + unsafe_rate + overdose_penalty;
}

extern "C" void kernel_launch(void* const* d_in, const int* in_sizes, int n_in,
                              void* d_out, int out_size, void* d_ws, size_t ws_size,
                              hipStream_t stream) {
    const float* x  = (const float*)d_in[0];   // 192 floats (32 seeds x 6)
    const float* rv = (const float*)d_in[1];   // 160^3 floats
    const float* om = (const float*)d_in[2];   // 160^3 floats
    float* out = (float*)d_out;
    float* ws  = (float*)d_ws;
    float* acc  = ws;        // 5 accumulators (+ pad)
    float* Bmat = ws + 16;   // 12x32 coefficient matrix

    dose_init<<<1, 32, 0, stream>>>(acc);
    dose_seed_prep<<<1, 32, 0, stream>>>(x, Bmat);
    dose_main<<<2048, 256, 0, stream>>>(rv, om, Bmat, acc);
    dose_finalize<<<1, 1, 0, stream>>>(acc, out);
}